// Total_loss_76802605187142
// MI455X (gfx1250) — compile-verified
//
#include <hip/hip_runtime.h>
#include <hip/hip_bf16.h>
#include <stdint.h>

// ---------------------------------------------------------------------------
// Problem constants (fixed by the reference setup_inputs):
//   B=64, T=2048, C=200 (cas has C+1=201 channels), D=256 (features: UNUSED).
// Memory-bound streaming reduction: ~211 MB read -> ~9 us floor @ 23.3 TB/s.
// Data path: gfx1250 async global->LDS loads (ASYNCcnt) double-buffer one
// (b,t) row per wave while the previous row is reduced. Math path: hardware
// v_log_f32 (log2 domain), with the common -ln2 factor applied once per wave
// so per-element cost is ~15 VALU + 2 trans + 2 DS  -> VALU time < HBM time.
// ---------------------------------------------------------------------------
constexpr int Bc  = 64;
constexpr int Tc  = 2048;
constexpr int Cc  = 200;
constexpr int Cp1 = 201;

#define WPB 8              // waves per block (256 threads)
#define ROWS_PER_WAVE 8    // rows (b,t) handled by each wave
constexpr int PA_OFF = 208;   // point_anno offset inside the LDS row buffer
constexpr int BUFSZ  = 416;   // 201 cas + pad + 200 anno, padded

constexpr float LN2F = 0.69314718055994530942f;

// ---- gfx1250 async global->LDS copy (ASYNCcnt path) -----------------------
#if __has_builtin(__builtin_amdgcn_global_load_async_to_lds_b32)
#define ASYNC_BUILTIN 1
#endif

typedef __attribute__((address_space(1))) int as1_int;   // global
typedef __attribute__((address_space(3))) int as3_int;   // LDS

__device__ __forceinline__ void async_cp_b32(const float* g, float* l) {
#ifdef ASYNC_BUILTIN
  __builtin_amdgcn_global_load_async_to_lds_b32(
      (as1_int*)(uintptr_t)g,
      (as3_int*)(uint32_t)(uintptr_t)l,   // low 32 bits of flat addr = LDS offset
      0, 0);
#else
  uint32_t loff = (uint32_t)(uintptr_t)l;
  asm volatile("global_load_async_to_lds_b32 %0, %1, off"
               :: "v"(loff), "v"(g)
               : "memory");
#endif
}

#if __has_builtin(__builtin_amdgcn_s_wait_asynccnt)
#define WAIT_ASYNC(N) __builtin_amdgcn_s_wait_asynccnt(N)
#else
#define WAIT_ASYNC(N) asm volatile("s_wait_asynccnt %0" :: "i"(N) : "memory")
#endif

// ---- helpers --------------------------------------------------------------
__device__ __forceinline__ float waveSum(float v) {
  for (int m = 16; m > 0; m >>= 1) v += __shfl_xor(v, m, 32);
  return v;
}

// ws layout (floats): [0,B) accAct | [B,2B) accBkg | [2B,3B) numAct
//                     [3B,4B) numBkg | [4B] vidSum
__global__ void zero_ws_kernel(float* ws, int n) {
  for (int i = threadIdx.x; i < n; i += blockDim.x) ws[i] = 0.0f;
}

// loss_vid: mean BCE over B*C = 12800 elements (tiny; keep accurate libm)
__global__ void vid_loss_kernel(const float* __restrict__ vs,
                                const float* __restrict__ lb,
                                float* __restrict__ acc, int n) {
  float s = 0.0f;
  for (int i = threadIdx.x; i < n; i += blockDim.x) {
    float p = vs[i], t = lb[i];
    float pc = fminf(fmaxf(p, 1e-7f), 1.0f - 1e-7f);
    s += -(t * logf(pc) + (1.0f - t) * log1pf(-pc));
  }
  s = waveSum(s);
  __shared__ float red[WPB];
  int lane = threadIdx.x & 31, w = threadIdx.x >> 5;
  if (lane == 0) red[w] = s;
  __syncthreads();
  if (threadIdx.x == 0) {
    float tot = 0.0f;
    for (int i = 0; i < (int)(blockDim.x >> 5); ++i) tot += red[i];
    atomicAdd(acc, tot);
  }
}

// Main streaming kernel: one wave per (b,t) row, async-staged double buffer.
__global__ __launch_bounds__(WPB * 32)
void frame_loss_kernel(const float* __restrict__ cas,
                       const float* __restrict__ pa,
                       const float* __restrict__ bkg,
                       float* __restrict__ ws) {
  __shared__ float lbuf[WPB][2][BUFSZ];
  const int lane = threadIdx.x & 31;
  const int w    = threadIdx.x >> 5;
  constexpr int tilesPerB = Tc / (WPB * ROWS_PER_WAVE);   // 32
  const int b    = blockIdx.x / tilesPerB;
  const int tile = blockIdx.x % tilesPerB;
  const int t0   = tile * (WPB * ROWS_PER_WAVE) + w * ROWS_PER_WAVE;
  const size_t rowBase = (size_t)b * Tc;

  auto stage = [&](int i) {
    const int t = t0 + i;
    const float* casRow = cas + (rowBase + t) * (size_t)Cp1;
    const float* paRow  = pa  + (rowBase + t) * (size_t)Cc;
    float* dst = &lbuf[w][i & 1][0];
#pragma unroll
    for (int k = 0; k < 7; ++k) {                    // 201 cas floats
      int c = lane + 32 * k;
      if (c < Cp1) async_cp_b32(casRow + c, dst + c);
    }
#pragma unroll
    for (int k = 0; k < 7; ++k) {                    // 200 anno floats
      int c = lane + 32 * k;
      if (c < Cc) async_cp_b32(paRow + c, dst + PA_OFF + c);
    }
  };  // exactly 14 async instructions per stage

  // Accumulated in log2 domain; scaled by -ln2 once before the atomics.
  float accAct = 0.0f, accBkg = 0.0f, nAct = 0.0f, nBkg = 0.0f;
  stage(0);
  for (int i = 0; i < ROWS_PER_WAVE; ++i) {
    if (i + 1 < ROWS_PER_WAVE) { stage(i + 1); WAIT_ASYNC(14); }  // row i done,
    else                       { WAIT_ASYNC(0); }                 // i+1 in flight
    const float* buf = &lbuf[w][i & 1][0];
    float aSum = 0.0f, bSum = 0.0f;
    bool  anyT = false;
#pragma unroll
    for (int k = 0; k < 7; ++k) {
      int c = lane + 32 * k;
      if (c < Cp1) {
        float p   = buf[c];
        float pc  = fminf(fmaxf(p, 1e-7f), 1.0f - 1e-7f);
        float lp  = __log2f(pc);            // v_log_f32 (trans pipe)
        float l1p = __log2f(1.0f - pc);     // exact 1-pc (Sterbenz) -> log2
        if (c < Cc) {
          float tv = buf[PA_OFF + c];                   // t in {0,1}
          float m  = fmaf(tv, fmaf(-2.0f, p, 1.0f), p); // (1-p)t + p(1-t)
          float inner = fmaf(tv, lp - l1p, l1p);        // t*lp + (1-t)*l1p
          aSum = fmaf(m * m, inner, aSum);              // (sign folded later)
          bSum = fmaf(p * p, l1p, bSum);                // bkg target = 0 here
          anyT = anyT || (tv != 0.0f);
        } else {                                        // c == 200
          aSum = fmaf(p * p, l1p, aSum);                // act target = 0
          float q = 1.0f - p;
          bSum = fmaf(q * q, lp, bSum);                 // bkg target = 1
        }
      }
    }
    aSum = waveSum(aSum);
    bSum = waveSum(bSum);
    float wmax = __any(anyT) ? 1.0f : 0.0f;   // max_c(t) with t in {0,1}
    float wb   = bkg[rowBase + (size_t)(t0 + i)];
    accAct += wmax * aSum;
    accBkg += wb   * bSum;
    nAct   += wmax;
    nBkg   += wb;
  }
  if (lane == 0) {
    // bce = -ln2 * (log2-domain inner sums)
    atomicAdd(&ws[0 * Bc + b], accAct * (-LN2F));
    atomicAdd(&ws[1 * Bc + b], accBkg * (-LN2F));
    atomicAdd(&ws[2 * Bc + b], nAct);
    atomicAdd(&ws[3 * Bc + b], nBkg);
  }
}

__global__ void finalize_kernel(const float* __restrict__ ws,
                                float* __restrict__ out) {
  __shared__ float sA[Bc], sB[Bc];
  int b = threadIdx.x;
  if (b < Bc) {
    sA[b] = ws[0 * Bc + b] / ws[2 * Bc + b];
    sB[b] = ws[1 * Bc + b] / ws[3 * Bc + b];
  }
  __syncthreads();
  if (threadIdx.x == 0) {
    float la = 0.0f, lb2 = 0.0f;
    for (int i = 0; i < Bc; ++i) { la += sA[i]; lb2 += sB[i]; }
    float loss_vid = ws[4 * Bc] / (float)(Bc * Cc);
    // loss_score == 0, loss_feat == 0 in the reference
    out[0] = loss_vid + la / (float)Bc + lb2 / (float)Bc;
  }
}

extern "C" void kernel_launch(void* const* d_in, const int* in_sizes, int n_in,
                              void* d_out, int out_size, void* d_ws, size_t ws_size,
                              hipStream_t stream) {
  const float* vid_score = (const float*)d_in[0];
  const float* cas       = (const float*)d_in[1];
  // d_in[2] = features — unused by the reference loss (loss_feat == 0)
  const float* label     = (const float*)d_in[3];
  const float* pa        = (const float*)d_in[4];
  const float* bkg       = (const float*)d_in[5];
  // d_in[6] = step — unused
  (void)in_sizes; (void)n_in; (void)out_size; (void)ws_size;

  float* ws  = (float*)d_ws;
  float* out = (float*)d_out;

  zero_ws_kernel<<<1, 256, 0, stream>>>(ws, 4 * Bc + 1);
  vid_loss_kernel<<<1, 256, 0, stream>>>(vid_score, label, ws + 4 * Bc, Bc * Cc);
  constexpr int blocks = Bc * (Tc / (WPB * ROWS_PER_WAVE));   // 2048
  frame_loss_kernel<<<blocks, WPB * 32, 0, stream>>>(cas, pa, bkg, ws);
  finalize_kernel<<<1, 64, 0, stream>>>(ws, out);
}